// MWF_43215960932438
// MI455X (gfx1250) — compile-verified
//
#include <hip/hip_runtime.h>
#include <math.h>

// Problem constants (from reference): B=2, S=4, C=2, F=2049, T=512
#define NB 2
#define NS 4
#define NC 2
#define NF 2049
#define NT 512

#define EPSF      1.1920928955078125e-07f   // np.finfo(f32).eps
#define SQRT_EPSF 3.4526698300124393e-04f   // sqrt(eps)

typedef float v2f __attribute__((ext_vector_type(2)));
typedef float v8f __attribute__((ext_vector_type(8)));

#define YSTR 516   // 512 + 4 pad -> conflict-free 16-row column gather

// ---------------- Kernel 1: global max of |X|^2 ----------------
__global__ __launch_bounds__(256)
void MWF_maxabs2_kernel(const float* __restrict__ re, const float* __restrict__ im,
                        unsigned int* __restrict__ wsmax, int n) {
    __shared__ float red[256];
    float m = 0.0f;
    for (int i = blockIdx.x * blockDim.x + threadIdx.x; i < n; i += gridDim.x * blockDim.x) {
        float r = re[i], q = im[i];
        m = fmaxf(m, r * r + q * q);
    }
    red[threadIdx.x] = m;
    __syncthreads();
    for (int s = 128; s > 0; s >>= 1) {
        if (threadIdx.x < (unsigned)s) red[threadIdx.x] = fmaxf(red[threadIdx.x], red[threadIdx.x + s]);
        __syncthreads();
    }
    if (threadIdx.x == 0) atomicMax(wsmax, __float_as_uint(red[0]));  // bits monotonic for x>=0
}

// ---------------- Kernel 2: one workgroup per (b,f) bin ----------------
__global__ __launch_bounds__(256)
void MWF_main_kernel(const float* __restrict__ msk, const float* __restrict__ mixr,
                     const float* __restrict__ mixi, const unsigned int* __restrict__ wsmax,
                     float* __restrict__ out) {
    // LDS: 33024 + 10240 + 8192 + 8192 + 1024 + 128 = 60800 bytes
    __shared__ float  Y[16 * YSTR];    // 16 rows = (s,c,{re,im}) x T, padded stride
    __shared__ float  vsh[NT * 5];     // v[t*5+s] (stride 5: conflict-free)
    __shared__ float4 xsh[NT];         // scaled mixture x[t] = (x0r,x0i,x1r,x1i)
    __shared__ float  Gw[8 * 256];     // per-wave partial Gram matrices
    __shared__ float  Gf[256];         // reduced 16x16 Gram
    __shared__ float  Rsh[NS * 8];     // per-source normalized R (complex 2x2)

    const int tid = threadIdx.x;
    const int bf  = blockIdx.x;
    const int b   = bf / NF;
    const int f   = bf % NF;

    const float max_abs = fmaxf(1.0f, sqrtf(__uint_as_float(*wsmax)) * 0.1f);
    const float inv_max = 1.0f / max_abs;

    const size_t mixbase = ((size_t)(b * NC)) * NF * NT + (size_t)f * NT;      // +c*NF*NT + t
    const size_t mskbase = ((size_t)(b * NS)) * NC * NF * NT + (size_t)f * NT; // +(s*NC+c)*NF*NT + t

    // ---- Pass 1: build Y rows, v[t,s], x[t] (single coalesced read of inputs) ----
    for (int t = tid; t < NT; t += 256) {
        float phr[NC], phi[NC], mg[NC];
        float xr[NC], xi[NC];
#pragma unroll
        for (int c = 0; c < NC; ++c) {
            float r = mixr[mixbase + (size_t)c * NF * NT + t];
            float q = mixi[mixbase + (size_t)c * NF * NT + t];
            float m = sqrtf(r * r + q * q);
            float inv = 1.0f / (m + EPSF);
            phr[c] = r * inv;  phi[c] = q * inv;  mg[c] = m;
            xr[c] = r * inv_max;  xi[c] = q * inv_max;
        }
        xsh[t] = make_float4(xr[0], xi[0], xr[1], xi[1]);
#pragma unroll
        for (int s = 0; s < NS; ++s) {
            float vacc = 0.0f;
#pragma unroll
            for (int c = 0; c < NC; ++c) {
                float a  = msk[mskbase + (size_t)(s * NC + c) * NF * NT + t] * mg[c] * inv_max;
                float yr = a * phr[c];
                float yi = a * phi[c];
                Y[(s * 4 + c * 2 + 0) * YSTR + t] = yr;
                Y[(s * 4 + c * 2 + 1) * YSTR + t] = yi;
                vacc += yr * yr + yi * yi;
            }
            vsh[t * 5 + s] = 0.5f * vacc;   // mean over C=2 channels
        }
    }
    __syncthreads();

    // ---- Pass 2: G = Y * Y^T via V_WMMA_F32_16X16X4_F32 (A layout == B layout for Y^T) ----
    const int lane  = tid & 31;
    const int wave  = tid >> 5;
    const int row   = lane & 15;
    const int thalf = (lane >> 4) << 1;   // lanes 16..31 carry K+2,K+3
    v8f G = {};
#pragma unroll
    for (int i = 0; i < 16; ++i) {        // 8 waves x 16 chunks x K=4 = T=512
        int t0 = (wave * 16 + i) * 4 + thalf;
        v2f a;
        a.x = Y[row * YSTR + t0];
        a.y = Y[row * YSTR + t0 + 1];
        G = __builtin_amdgcn_wmma_f32_16x16x4_f32(false, a, false, a, (short)0, G, false, false);
    }
    {   // scatter wave-partial G to LDS: VGPR i, lane l -> (M = i + 8*(l>>4), N = l&15)
        int n = lane & 15;
        int mbase = (lane >> 4) << 3;
#pragma unroll
        for (int i = 0; i < 8; ++i) Gw[wave * 256 + (mbase + i) * 16 + n] = G[i];
    }
    __syncthreads();
    {   // reduce 8 wave-partials
        float acc = 0.0f;
#pragma unroll
        for (int w = 0; w < 8; ++w) acc += Gw[w * 256 + tid];
        Gf[tid] = acc;
    }
    __syncthreads();

    // ---- R[s] and weight from Gram matrix ----
    if (tid < NS) {
        int s = tid, r0 = s * 4;          // rows: r0+0 c0re, +1 c0im, +2 c1re, +3 c1im
        float g00 = Gf[(r0 + 0) * 16 + (r0 + 0)];
        float g11 = Gf[(r0 + 1) * 16 + (r0 + 1)];
        float g22 = Gf[(r0 + 2) * 16 + (r0 + 2)];
        float g33 = Gf[(r0 + 3) * 16 + (r0 + 3)];
        float invw = 1.0f / (EPSF + 0.5f * (g00 + g11 + g22 + g33));
        float R00r = (g00 + g11) * invw;
        float R11r = (g22 + g33) * invw;
        float R01r = (Gf[(r0 + 0) * 16 + (r0 + 2)] + Gf[(r0 + 1) * 16 + (r0 + 3)]) * invw;
        float R01i = (Gf[(r0 + 1) * 16 + (r0 + 2)] - Gf[(r0 + 0) * 16 + (r0 + 3)]) * invw;
        Rsh[s * 8 + 0] = R00r;  Rsh[s * 8 + 1] = 0.0f;
        Rsh[s * 8 + 2] = R01r;  Rsh[s * 8 + 3] = R01i;
        Rsh[s * 8 + 4] = R01r;  Rsh[s * 8 + 5] = -R01i;
        Rsh[s * 8 + 6] = R11r;  Rsh[s * 8 + 7] = 0.0f;
    }
    __syncthreads();

    // ---- Pass 3: per-t Wiener solve + coalesced complex64 stores ----
    for (int t = tid; t < NT; t += 256) {
        float4 x4 = xsh[t];
        float vv[NS];
#pragma unroll
        for (int s = 0; s < NS; ++s) vv[s] = vsh[t * 5 + s];

        float c00r = SQRT_EPSF, c00i = 0.f, c01r = 0.f, c01i = 0.f;
        float c10r = 0.f, c10i = 0.f, c11r = SQRT_EPSF, c11i = 0.f;
#pragma unroll
        for (int s = 0; s < NS; ++s) {
            float v = vv[s];
            c00r += v * Rsh[s * 8 + 0];  c00i += v * Rsh[s * 8 + 1];
            c01r += v * Rsh[s * 8 + 2];  c01i += v * Rsh[s * 8 + 3];
            c10r += v * Rsh[s * 8 + 4];  c10i += v * Rsh[s * 8 + 5];
            c11r += v * Rsh[s * 8 + 6];  c11i += v * Rsh[s * 8 + 7];
        }
        // det = c00*c11 - c01*c10 ; invdet = conj(det)/|det|^2
        float detr = c00r * c11r - c00i * c11i - (c01r * c10r - c01i * c10i);
        float deti = c00r * c11i + c00i * c11r - (c01r * c10i + c01i * c10r);
        float dmag = detr * detr + deti * deti;
        float idr = detr / dmag, idi = -deti / dmag;
        float i00r =  (c11r * idr - c11i * idi), i00i =  (c11r * idi + c11i * idr);
        float i01r = -(c01r * idr - c01i * idi), i01i = -(c01r * idi + c01i * idr);
        float i10r = -(c10r * idr - c10i * idi), i10i = -(c10r * idi + c10i * idr);
        float i11r =  (c00r * idr - c00i * idi), i11i =  (c00r * idi + c00i * idr);
        // z = inv(Cxx) @ x
        float x0r = x4.x, x0i = x4.y, x1r = x4.z, x1i = x4.w;
        float z0r = i00r * x0r - i00i * x0i + i01r * x1r - i01i * x1i;
        float z0i = i00r * x0i + i00i * x0r + i01r * x1i + i01i * x1r;
        float z1r = i10r * x0r - i10i * x0i + i11r * x1r - i11i * x1i;
        float z1i = i10r * x0i + i10i * x0r + i11r * x1i + i11i * x1r;
#pragma unroll
        for (int s = 0; s < NS; ++s) {
            float v = vv[s] * max_abs;
            float R00r = Rsh[s * 8 + 0], R00i = Rsh[s * 8 + 1];
            float R01r = Rsh[s * 8 + 2], R01i = Rsh[s * 8 + 3];
            float R10r = Rsh[s * 8 + 4], R10i = Rsh[s * 8 + 5];
            float R11r = Rsh[s * 8 + 6], R11i = Rsh[s * 8 + 7];
            float y0r = v * (R00r * z0r - R00i * z0i + R01r * z1r - R01i * z1i);
            float y0i = v * (R00r * z0i + R00i * z0r + R01r * z1i + R01i * z1r);
            float y1r = v * (R10r * z0r - R10i * z0i + R11r * z1r - R11i * z1i);
            float y1i = v * (R10r * z0i + R10i * z0r + R11r * z1i + R11i * z1r);
            size_t o0 = ((((size_t)((b * NS + s) * NC + 0)) * NF + f) * NT + t) * 2;
            size_t o1 = ((((size_t)((b * NS + s) * NC + 1)) * NF + f) * NT + t) * 2;
            *(float2*)(out + o0) = make_float2(y0r, y0i);
            *(float2*)(out + o1) = make_float2(y1r, y1i);
        }
    }
}

extern "C" void kernel_launch(void* const* d_in, const int* in_sizes, int n_in,
                              void* d_out, int out_size, void* d_ws, size_t ws_size,
                              hipStream_t stream) {
    const float* msk  = (const float*)d_in[0];   // (B,S,C,F,T)
    const float* mixr = (const float*)d_in[1];   // (B,C,F,T)
    const float* mixi = (const float*)d_in[2];   // (B,C,F,T)
    float* out = (float*)d_out;                  // (B,S,C,F,T) complex64 interleaved
    unsigned int* wsmax = (unsigned int*)d_ws;

    hipMemsetAsync(d_ws, 0, sizeof(unsigned int), stream);
    const int n = NB * NC * NF * NT;
    MWF_maxabs2_kernel<<<1024, 256, 0, stream>>>(mixr, mixi, wsmax, n);
    MWF_main_kernel<<<NB * NF, 256, 0, stream>>>(msk, mixr, mixi, wsmax, out);
}